// LigerLMHeadPreference_67920612819621
// MI455X (gfx1250) — compile-verified
//
#include <hip/hip_runtime.h>
#include <hip/hip_bf16.h>
#include <math.h>
#include <stdint.h>

// Problem constants (from reference)
#define HDIM 4096
#define VOCAB 32000
#define NTOK 4096      // 8 * 512 tokens
#define TSEQ 512
#define NSEQ 8
#define BETA 0.1f
#define IGNORE_INDEX (-100)

// Tiling
#define MBLK 64               // tokens per block
#define NBLK 128              // vocab per block
#define KC   32               // K chunk staged in LDS
#define KCP  36               // padded K stride (TDM pad: 4 dwords per 32 dwords)
#define CSP  132              // padded C stride
#define NVB  (VOCAB / NBLK)   // 250 vocab blocks
#define NTB  (NTOK / MBLK)    // 64 token blocks
#define NCHUNK (HDIM / KC)    // 128

// LDS layout (floats): [A0 | A1 | B0 | B1], Cs aliased on top afterwards
#define A_FLOATS (MBLK * KCP)          // 2304
#define B_FLOATS (NBLK * KCP)          // 4608
#define AOFF(s)  ((s) * A_FLOATS)
#define BOFF(s)  (2 * A_FLOATS + (s) * B_FLOATS)
#define SMEM_FLOATS (2 * A_FLOATS + 2 * B_FLOATS)   // 13824 floats = 55.3 KB

typedef __attribute__((ext_vector_type(2))) float    v2f;
typedef __attribute__((ext_vector_type(8))) float    v8f;
typedef __attribute__((ext_vector_type(4))) uint32_t u32x4;
typedef __attribute__((ext_vector_type(8))) int      i32x8;
typedef __attribute__((ext_vector_type(4))) int      i32x4;

// ---------------------------------------------------------------------------
// Build a 2D TDM descriptor and issue TENSOR_LOAD_TO_LDS (6-arg builtin).
// Pads LDS rows: 4 DWORDs after every 32 DWORDs -> LDS row stride 36 floats.
// ---------------------------------------------------------------------------
__device__ __forceinline__ void tdm_load_tile(
    uint32_t lds_byte, const float* gsrc,
    uint32_t tile_cols, uint32_t tile_rows,
    uint32_t tensor_cols, uint32_t tensor_rows, uint32_t row_stride_elems)
{
    uint64_t ga = (uint64_t)(uintptr_t)gsrc;
    u32x4 g0;
    g0.x = 1u;                                           // count=1, user D#
    g0.y = lds_byte;                                     // lds_addr
    g0.z = (uint32_t)ga;                                 // global_addr[31:0]
    g0.w = (uint32_t)((ga >> 32) & 0x01FFFFFFu)          // global_addr[56:32]
         | (2u << 30);                                   // type = 2 ("image")
    i32x8 g1;
    g1[0] = (2 << 16)                                    // data_size = 4 bytes
          | (1 << 20)                                    // pad_enable
          | (4 << 22)                                    // pad_interval: 32 DW
          | (3 << 25);                                   // pad_amount: 4 DW
    g1[1] = (int)((tensor_cols & 0xFFFFu) << 16);        // tensor_dim0 lo16
    g1[2] = (int)((tensor_cols >> 16)
          | ((tensor_rows & 0xFFFFu) << 16));            // dim0 hi16 | dim1 lo16
    g1[3] = (int)((tensor_rows >> 16)
          | (tile_cols << 16));                          // dim1 hi16 | tile_dim0
    g1[4] = (int)tile_rows;                              // tile_dim1 (tile_dim2=0)
    g1[5] = (int)row_stride_elems;                       // tensor_dim0_stride lo32
    g1[6] = 0;
    g1[7] = 0;
    i32x4 z4 = {0, 0, 0, 0};
    i32x8 z8 = {0, 0, 0, 0, 0, 0, 0, 0};
    __builtin_amdgcn_tensor_load_to_lds(g0, g1, z4, z4, z8, 0);
}

// ---------------------------------------------------------------------------
// Kernel 1: fused GEMM (WMMA f32 16x16x4) + per-vocab-slab online softmax
// partials. TDM-fed double-buffered LDS. grid.x = vBlk * NTB + tBlk
// (token-minor => each 2MB W slab is reused across 64 token blocks via L2)
// ---------------------------------------------------------------------------
__global__ __launch_bounds__(256) void dpo_gemm_partials(
    const float* __restrict__ x, const float* __restrict__ w,
    const int* __restrict__ target,
    float* __restrict__ pmax, float* __restrict__ psum,
    float* __restrict__ tlogit)
{
    __shared__ __align__(16) float smem[SMEM_FLOATS];

    const int bx      = blockIdx.x;
    const int vBlk    = bx / NTB;
    const int tBlk    = bx % NTB;
    const int vocBase = vBlk * NBLK;
    const int tokBase = tBlk * MBLK;

    const int tid  = threadIdx.x;
    const int lane = tid & 31;
    const int wave = tid >> 5;
    const int wm   = wave >> 1;      // 0..3  (wave row: 16 tokens)
    const int wn   = wave & 1;       // 0..1  (wave col: 64 vocab)
    const int half = lane >> 4;      // lane half selects K split / M split
    const int ln   = lane & 15;

    // LDS byte addresses for TDM (low 32 bits of generic ptr == LDS offset)
    const uint32_t ldsA0 = (uint32_t)(uintptr_t)&smem[AOFF(0)];
    const uint32_t ldsA1 = (uint32_t)(uintptr_t)&smem[AOFF(1)];
    const uint32_t ldsB0 = (uint32_t)(uintptr_t)&smem[BOFF(0)];
    const uint32_t ldsB1 = (uint32_t)(uintptr_t)&smem[BOFF(1)];

    v8f acc[4];
    #pragma unroll
    for (int t = 0; t < 4; ++t)
        #pragma unroll
        for (int j = 0; j < 8; ++j) acc[t][j] = 0.0f;

    // Prologue: stage chunk 0 into buffer 0 (one wave drives the TDM)
    if (wave == 0) {
        tdm_load_tile(ldsA0, &x[(size_t)tokBase * HDIM], KC, MBLK,
                      HDIM, (uint32_t)(NTOK - tokBase), HDIM);
        tdm_load_tile(ldsB0, &w[(size_t)vocBase * HDIM], KC, NBLK,
                      HDIM, (uint32_t)(VOCAB - vocBase), HDIM);
    }

    for (int i = 0; i < NCHUNK; ++i) {
        const int cur = i & 1;
        if (wave == 0) {
            if (i + 1 < NCHUNK) {
                const int k1 = (i + 1) * KC;
                tdm_load_tile(cur ? ldsA0 : ldsA1,
                              &x[(size_t)tokBase * HDIM + k1], KC, MBLK,
                              HDIM, (uint32_t)(NTOK - tokBase), HDIM);
                tdm_load_tile(cur ? ldsB0 : ldsB1,
                              &w[(size_t)vocBase * HDIM + k1], KC, NBLK,
                              HDIM, (uint32_t)(VOCAB - vocBase), HDIM);
                // 2 TDM ops per chunk; <=2 outstanding => chunk i is complete
                __builtin_amdgcn_s_wait_tensorcnt(2);
            } else {
                __builtin_amdgcn_s_wait_tensorcnt(0);
            }
        }
        __syncthreads();   // publish buf[cur]; also fences prior readers

        const float* As = &smem[AOFF(cur)];
        const float* Bs = &smem[BOFF(cur)];

        #pragma unroll
        for (int kk = 0; kk < KC; kk += 4) {
            const int ko = kk + half * 2;  // lanes 0-15: K=kk,kk+1 ; 16-31: +2,+3
            v2f a = *(const v2f*)&As[(wm * 16 + ln) * KCP + ko];
            #pragma unroll
            for (int t = 0; t < 4; ++t) {
                v2f b = *(const v2f*)&Bs[(wn * 64 + t * 16 + ln) * KCP + ko];
                acc[t] = __builtin_amdgcn_wmma_f32_16x16x4_f32(
                    false, a, false, b, (short)0, acc[t], false, false);
            }
        }
        __syncthreads();   // all reads of buf[cur] done before it is re-filled
    }

    // Alias logits tile over the (now idle) staging buffers
    float* Cs = smem;      // needs 64*132 = 8448 floats <= 13824

    // C layout: VGPR j: lanes 0-15 -> M=j, lanes 16-31 -> M=8+j; N = ln
    #pragma unroll
    for (int t = 0; t < 4; ++t)
        #pragma unroll
        for (int j = 0; j < 8; ++j) {
            int row = wm * 16 + half * 8 + j;
            int col = wn * 64 + t * 16 + ln;
            Cs[row * CSP + col] = acc[t][j];
        }
    __syncthreads();

    // Per-token slab partials: max, sum(exp(v - max)), target logit
    if (tid < MBLK) {
        const int r   = tid;
        const int tok = tokBase + r;
        float m = -INFINITY;
        for (int c = 0; c < NBLK; ++c) m = fmaxf(m, Cs[r * CSP + c]);
        float s = 0.f;
        for (int c = 0; c < NBLK; ++c) s += expf(Cs[r * CSP + c] - m);
        pmax[(size_t)vBlk * NTOK + tok] = m;
        psum[(size_t)vBlk * NTOK + tok] = s;
        int tg = target[tok];
        if (tg != IGNORE_INDEX && tg >= vocBase && tg < vocBase + NBLK)
            tlogit[tok] = Cs[r * CSP + (tg - vocBase)];
    }
}

// ---------------------------------------------------------------------------
// Kernel 2: combine slab partials -> per-token log-prob at target
// ---------------------------------------------------------------------------
__global__ __launch_bounds__(256) void dpo_combine(
    const int* __restrict__ target,
    const float* __restrict__ pmax, const float* __restrict__ psum,
    const float* __restrict__ tlogit,
    float* __restrict__ logp, float* __restrict__ maskf)
{
    int tok = blockIdx.x * blockDim.x + threadIdx.x;
    if (tok >= NTOK) return;
    int tg = target[tok];
    if (tg == IGNORE_INDEX) { logp[tok] = 0.f; maskf[tok] = 0.f; return; }
    float m = -INFINITY;
    for (int sb = 0; sb < NVB; ++sb)
        m = fmaxf(m, pmax[(size_t)sb * NTOK + tok]);
    float s = 0.f;
    for (int sb = 0; sb < NVB; ++sb)
        s += psum[(size_t)sb * NTOK + tok] * expf(pmax[(size_t)sb * NTOK + tok] - m);
    logp[tok]  = tlogit[tok] - (m + logf(s));
    maskf[tok] = 1.f;
}

// ---------------------------------------------------------------------------
// Kernel 3: masked per-sequence mean, DPO pairing, final scalar
// ---------------------------------------------------------------------------
__global__ __launch_bounds__(256) void dpo_final(
    const float* __restrict__ logp, const float* __restrict__ maskf,
    float* __restrict__ out)
{
    __shared__ float red[256];
    __shared__ float avg[NSEQ];
    const int tid = threadIdx.x;

    for (int b = 0; b < NSEQ; ++b) {
        float sl = 0.f, sm = 0.f;
        for (int i = tid; i < TSEQ; i += 256) {
            sl += logp[b * TSEQ + i];
            sm += maskf[b * TSEQ + i];
        }
        red[tid] = sl; __syncthreads();
        for (int st = 128; st > 0; st >>= 1) {
            if (tid < st) red[tid] += red[tid + st];
            __syncthreads();
        }
        float tot_l = red[0]; __syncthreads();
        red[tid] = sm; __syncthreads();
        for (int st = 128; st > 0; st >>= 1) {
            if (tid < st) red[tid] += red[tid + st];
            __syncthreads();
        }
        if (tid == 0) avg[b] = tot_l / fmaxf(red[0], 1.f);
        __syncthreads();
    }
    if (tid == 0) {
        float a = 0.f;
        for (int i = 0; i < NSEQ / 2; ++i) a += avg[i] - avg[i + NSEQ / 2];
        out[0] = -BETA * (a / (float)(NSEQ / 2));
    }
}

// ---------------------------------------------------------------------------
extern "C" void kernel_launch(void* const* d_in, const int* in_sizes, int n_in,
                              void* d_out, int out_size, void* d_ws, size_t ws_size,
                              hipStream_t stream) {
    const float* x      = (const float*)d_in[0];   // (8,512,4096) f32
    const float* w      = (const float*)d_in[1];   // (32000,4096) f32
    const int*   target = (const int*)d_in[2];     // (8,512)

    float* pmax   = (float*)d_ws;                        // [NVB][NTOK]
    float* psum   = pmax   + (size_t)NVB * NTOK;         // [NVB][NTOK]
    float* tlogit = psum   + (size_t)NVB * NTOK;         // [NTOK]
    float* logp   = tlogit + NTOK;                       // [NTOK]
    float* maskf  = logp   + NTOK;                       // [NTOK]

    dpo_gemm_partials<<<NVB * NTB, 256, 0, stream>>>(x, w, target, pmax, psum, tlogit);
    dpo_combine<<<NTOK / 256, 256, 0, stream>>>(target, pmax, psum, tlogit, logp, maskf);
    dpo_final<<<1, 256, 0, stream>>>(logp, maskf, (float*)d_out);
}